// LDDMM_44933947851013
// MI455X (gfx1250) — compile-verified
//
#include <hip/hip_runtime.h>
#include <hip/hip_bf16.h>

#define HH 1024
#define WW 1024
#define HWSZ ((size_t)HH * (size_t)WW)

typedef float v2f __attribute__((ext_vector_type(2)));
typedef float v8f __attribute__((ext_vector_type(8)));

#if defined(__has_builtin)
#if __has_builtin(__builtin_amdgcn_wmma_f32_16x16x4_f32)
#define HAVE_WMMA4 1
#endif
#endif

// Normalized 7-tap Gaussian, sigma=1 (matches jnp kernel to f32 rounding)
__device__ __constant__ float c_GK[7] = {
    0.004433048175f, 0.054005582622f, 0.242036229400f, 0.399050279700f,
    0.242036229400f, 0.054005582622f, 0.004433048175f};

__device__ __forceinline__ int refl(int i, int n) {
  if (i < 0) i = -i;
  if (i >= n) i = 2 * n - 2 - i;
  return i;
}

// band(j) = GK[j] if 0<=j<7 else 0  (banded blur matrix entry)
__device__ __forceinline__ float band(int j) {
  int jc = j < 0 ? 0 : (j > 6 ? 6 : j);
  float w = c_GK[jc];
  return ((unsigned)j < 7u) ? w : 0.0f;
}

// ---------------- Vertical blur: D = Band(16x24) * In(24x16) ----------------
// One wave per 16x16 output tile. blockDim=(32,4): 4 tiles per block.
__global__ __launch_bounds__(128) void vblur_wmma(const float* __restrict__ in,
                                                  float* __restrict__ out) {
  const int lane = threadIdx.x;
  const int lr = lane & 15;
  const int half = lane >> 4;
  const int tc = blockIdx.x * 16;
  const int tr = (blockIdx.y * blockDim.y + threadIdx.y) * 16;
  const float* ip = in + (size_t)blockIdx.z * HWSZ;
  float* op = out + (size_t)blockIdx.z * HWSZ;

  v8f d = {0.f, 0.f, 0.f, 0.f, 0.f, 0.f, 0.f, 0.f};
#if HAVE_WMMA4
#pragma unroll
  for (int kc = 0; kc < 6; ++kc) {
    const int k0 = 4 * kc + 2 * half;
    v2f a, b;
    // A[o][i] = kern[i-o], row o = lr, i = k0, k0+1
    a.x = band(k0 - lr);
    a.y = band(k0 + 1 - lr);
    // B[i][n] = in[tr-3+i][tc+n], col n = lr
    const int r0 = refl(tr - 3 + k0, HH);
    const int r1 = refl(tr - 3 + k0 + 1, HH);
    b.x = ip[(size_t)r0 * WW + tc + lr];
    b.y = ip[(size_t)r1 * WW + tc + lr];
    d = __builtin_amdgcn_wmma_f32_16x16x4_f32(false, a, false, b, (short)0, d,
                                              false, false);
  }
#pragma unroll
  for (int vv = 0; vv < 8; ++vv)
    op[(size_t)(tr + vv + 8 * half) * WW + tc + lr] = d[vv];
#else
#pragma unroll
  for (int vv = 0; vv < 8; ++vv) {
    const int row = tr + vv + 8 * half;
    float acc = 0.f;
#pragma unroll
    for (int j = 0; j < 7; ++j)
      acc += c_GK[j] * ip[(size_t)refl(row + j - 3, HH) * WW + tc + lr];
    op[(size_t)row * WW + tc + lr] = acc;
  }
#endif
}

// -------------- Horizontal blur: D = In(16x24) * Band(24x16) ---------------
__global__ __launch_bounds__(128) void hblur_wmma(const float* __restrict__ in,
                                                  float* __restrict__ out) {
  const int lane = threadIdx.x;
  const int lr = lane & 15;
  const int half = lane >> 4;
  const int tc = blockIdx.x * 16;
  const int tr = (blockIdx.y * blockDim.y + threadIdx.y) * 16;
  const float* ip = in + (size_t)blockIdx.z * HWSZ;
  float* op = out + (size_t)blockIdx.z * HWSZ;

  v8f d = {0.f, 0.f, 0.f, 0.f, 0.f, 0.f, 0.f, 0.f};
#if HAVE_WMMA4
#pragma unroll
  for (int kc = 0; kc < 6; ++kc) {
    const int k0 = 4 * kc + 2 * half;
    v2f a, b;
    // A[m][i] = in[tr+m][tc-3+i], row m = lr
    const int c0 = refl(tc - 3 + k0, WW);
    const int c1 = refl(tc - 3 + k0 + 1, WW);
    a.x = ip[(size_t)(tr + lr) * WW + c0];
    a.y = ip[(size_t)(tr + lr) * WW + c1];
    // B[i][n] = kern[i-n], col n = lr
    b.x = band(k0 - lr);
    b.y = band(k0 + 1 - lr);
    d = __builtin_amdgcn_wmma_f32_16x16x4_f32(false, a, false, b, (short)0, d,
                                              false, false);
  }
#pragma unroll
  for (int vv = 0; vv < 8; ++vv)
    op[(size_t)(tr + vv + 8 * half) * WW + tc + lr] = d[vv];
#else
#pragma unroll
  for (int vv = 0; vv < 8; ++vv) {
    const int row = tr + vv + 8 * half;
    float acc = 0.f;
#pragma unroll
    for (int j = 0; j < 7; ++j)
      acc += c_GK[j] * ip[(size_t)row * WW + refl(tc + lr + j - 3, WW)];
    op[(size_t)row * WW + tc + lr] = acc;
  }
#endif
}

// -------- Flow integration (batch-invariant: computed once, fanned x8) -----
__global__ __launch_bounds__(256) void integrate_flow(
    const float* __restrict__ vb, float* __restrict__ flow) {
  const size_t p = (size_t)blockIdx.x * blockDim.x + threadIdx.x;
  if (p >= HWSZ) return;
  const int y = (int)(p / WW);
  const int x = (int)(p % WW);
  float u = 0.f, v = 0.f;
#pragma unroll
  for (int t = 0; t < 5; ++t) {
    const float* vx = vb + (size_t)(2 * t) * HWSZ;
    const float* vy = vx + HWSZ;
    float px = fminf(fmaxf((float)x + u, 0.f), (float)(WW - 1));
    float py = fminf(fmaxf((float)y + v, 0.f), (float)(HH - 1));
    float x0f = floorf(px), y0f = floorf(py);
    int x0 = (int)x0f, y0 = (int)y0f;
    int x1 = min(x0 + 1, WW - 1), y1 = min(y0 + 1, HH - 1);
    float wx = px - x0f, wy = py - y0f;
    float w00 = (1.f - wx) * (1.f - wy), w10 = wx * (1.f - wy);
    float w01 = (1.f - wx) * wy, w11 = wx * wy;
    size_t i00 = (size_t)y0 * WW + x0, i10 = (size_t)y0 * WW + x1;
    size_t i01 = (size_t)y1 * WW + x0, i11 = (size_t)y1 * WW + x1;
    float su = vx[i00] * w00 + vx[i10] * w10 + vx[i01] * w01 + vx[i11] * w11;
    float sv = vy[i00] * w00 + vy[i10] * w10 + vy[i01] * w01 + vy[i11] * w11;
    if (t < 4) {  // pull next step's planes toward L2 (global_prefetch_b8)
      __builtin_prefetch(vx + 2 * HWSZ + i00, 0, 1);
      __builtin_prefetch(vy + 2 * HWSZ + i00, 0, 1);
    }
    u += 0.2f * su;
    v += 0.2f * sv;
  }
#pragma unroll
  for (int b = 0; b < 8; ++b) {
    flow[(size_t)b * 2 * HWSZ + p] = u;
    flow[(size_t)b * 2 * HWSZ + HWSZ + p] = v;
  }
}

// ------------- Warp source by flow, scale by Jacobian determinant ----------
__global__ __launch_bounds__(256) void warp_det(const float* __restrict__ src,
                                                const float* __restrict__ flow,
                                                float* __restrict__ warped) {
  const size_t idx = (size_t)blockIdx.x * blockDim.x + threadIdx.x;
  if (idx >= 8 * HWSZ) return;
  const int b = (int)(idx / HWSZ);
  const size_t p = idx % HWSZ;
  const int y = (int)(p / WW);
  const int x = (int)(p % WW);

  const float* fu = flow + (size_t)b * 2 * HWSZ;
  const float* fv = fu + HWSZ;
  const float u = fu[p];
  const float v = fv[p];

  const int xm = max(x - 1, 0), xp = min(x + 1, WW - 1);
  const int ym = max(y - 1, 0), yp = min(y + 1, HH - 1);
  const float rdx = 1.f / (float)(xp - xm);
  const float rdy = 1.f / (float)(yp - ym);
  const float du_dx = (fu[(size_t)y * WW + xp] - fu[(size_t)y * WW + xm]) * rdx;
  const float du_dy = (fu[(size_t)yp * WW + x] - fu[(size_t)ym * WW + x]) * rdy;
  const float dv_dx = (fv[(size_t)y * WW + xp] - fv[(size_t)y * WW + xm]) * rdx;
  const float dv_dy = (fv[(size_t)yp * WW + x] - fv[(size_t)ym * WW + x]) * rdy;
  const float det = (1.f + du_dx) * (1.f + dv_dy) - du_dy * dv_dx;

  const float* sp = src + (size_t)b * HWSZ;
  float px = fminf(fmaxf((float)x + u, 0.f), (float)(WW - 1));
  float py = fminf(fmaxf((float)y + v, 0.f), (float)(HH - 1));
  float x0f = floorf(px), y0f = floorf(py);
  int x0 = (int)x0f, y0 = (int)y0f;
  int x1 = min(x0 + 1, WW - 1), y1 = min(y0 + 1, HH - 1);
  float wx = px - x0f, wy = py - y0f;
  float s = sp[(size_t)y0 * WW + x0] * (1.f - wx) * (1.f - wy) +
            sp[(size_t)y0 * WW + x1] * wx * (1.f - wy) +
            sp[(size_t)y1 * WW + x0] * (1.f - wx) * wy +
            sp[(size_t)y1 * WW + x1] * wx * wy;
  warped[(size_t)b * HWSZ + p] = s * det;
}

extern "C" void kernel_launch(void* const* d_in, const int* in_sizes, int n_in,
                              void* d_out, int out_size, void* d_ws,
                              size_t ws_size, hipStream_t stream) {
  const float* source = (const float*)d_in[0];    // (8,1,1024,1024)
  const float* velocity = (const float*)d_in[1];  // (5,2,1024,1024)
  float* out = (float*)d_out;                     // warped[8*HW] ++ flow[16*HW]
  float* warped = out;
  float* flow = out + 8 * HWSZ;

  float* blurred = (float*)d_ws;        // 10*HW floats (40 MB)
  float* tmp = blurred + 10 * HWSZ;     // 2*HW floats (8 MB)

  const dim3 bb(32, 4, 1);   // 4 waves/block, one 16x16 tile per wave
  const dim3 gb(WW / 16, HH / 16 / 4, 2);
  for (int t = 0; t < 5; ++t) {
    vblur_wmma<<<gb, bb, 0, stream>>>(velocity + (size_t)(2 * t) * HWSZ, tmp);
    hblur_wmma<<<gb, bb, 0, stream>>>(tmp, blurred + (size_t)(2 * t) * HWSZ);
  }

  const int threads = 256;
  integrate_flow<<<(int)((HWSZ + threads - 1) / threads), threads, 0, stream>>>(
      blurred, flow);
  warp_det<<<(int)((8 * HWSZ + threads - 1) / threads), threads, 0, stream>>>(
      source, flow, warped);
}